// SoftKConv_19473381720430
// MI455X (gfx1250) — compile-verified
//
#include <hip/hip_runtime.h>
#include <math.h>

// ---------------------------------------------------------------------------
// SoftKConv for MI455X (gfx1250, wave32).
//   K1: x = feat @ W            via V_WMMA_F32_16X16X4_F32
//   K2: per-node top-32 select, gather, Gram (WMMA), distances, softmax, agg
// ---------------------------------------------------------------------------

typedef __attribute__((ext_vector_type(2))) float v2f;
typedef __attribute__((ext_vector_type(8))) float v8f;

#define NNODES 50000
#define DEG    48
#define DIN    128
#define DOUT   128
#define KSEL   32

static __device__ __forceinline__ v8f wmma_f32(v2f a, v2f b, v8f c) {
  // 8 args: (neg_a, A, neg_b, B, c_mod, C, reuse_a, reuse_b)
  return __builtin_amdgcn_wmma_f32_16x16x4_f32(false, a, false, b, (short)0, c,
                                               false, false);
}

// ---------------------------------------------------------------------------
// Kernel 1: x[n,128] = feat[n,128] @ W[128,128]   (f32 WMMA)
// grid = (n/16, 2), block = 128 (4 waves). Wave w owns col-tile blockIdx.y*4+w.
// A panel (16x128) staged in LDS; B fragments preloaded to registers (W is
// 64 KB, L2-resident across all blocks).
// ---------------------------------------------------------------------------
__global__ __launch_bounds__(128) void xform_gemm(const float* __restrict__ feat,
                                                  const float* __restrict__ W,
                                                  float* __restrict__ x) {
  __shared__ float As[16 * DIN];   // 8 KB

  const int lane    = threadIdx.x & 31;
  const int wave    = threadIdx.x >> 5;
  const int rowbase = blockIdx.x * 16;
  const int colbase = (blockIdx.y * 4 + wave) * 16;

  // Cooperative, coalesced A-panel load: 16 rows x 128 f32 = 512 float4.
  const float4* fsrc = (const float4*)(feat + (size_t)rowbase * DIN);
  float4*       adst = (float4*)As;
#pragma unroll
  for (int i = 0; i < 4; ++i)
    adst[threadIdx.x + i * 128] = fsrc[threadIdx.x + i * 128];
  __syncthreads();

  const int kg = (lane >> 4) << 1;        // K sub-offset per half-wave: 0 / 2
  const int nn = colbase + (lane & 15);   // this lane's N column

  // Preload B fragments for all 32 K-steps (64 f32 regs / lane).
  float b0[32], b1[32];
#pragma unroll
  for (int kk = 0; kk < 32; ++kk) {
    b0[kk] = W[(kk * 4 + kg + 0) * DOUT + nn];
    b1[kk] = W[(kk * 4 + kg + 1) * DOUT + nn];
  }

  v8f acc = {};
  const int arow = (lane & 15) * DIN + kg;
#pragma unroll
  for (int kk = 0; kk < 32; ++kk) {
    v2f a = *(const v2f*)&As[arow + kk * 4];   // 8B-aligned LDS read
    v2f b;
    b.x = b0[kk];
    b.y = b1[kk];
    acc = wmma_f32(a, b, acc);
  }

  // C/D layout: VGPR r -> row r (lanes 0-15) / row r+8 (lanes 16-31).
  const int mbase = rowbase + ((lane >> 4) << 3);
#pragma unroll
  for (int r = 0; r < 8; ++r)
    x[(size_t)(mbase + r) * DOUT + nn] = acc[r];
}

// ---------------------------------------------------------------------------
// Kernel 2: per-node pipeline. One wave32 per node, 2 nodes per 64-thr block.
// ---------------------------------------------------------------------------
struct NodeSmem {
  float xi[KSEL * DOUT];    // 16 KB gathered neighbor features
  float gram[KSEL * KSEL];  // 4 KB  Gram matrix
  float wbuf[64];           // candidate weights (49 used)
  int   nbbuf[64];          // candidate neighbor ids
  float selw[KSEL];         // top-32 weights
  int   selidx[KSEL];       // top-32 neighbor ids
  float distb[KSEL];        // per-row weighted distances
  float relb[KSEL];         // relevance weights
};                          // 21.5 KB/node -> 43 KB/block

__global__ __launch_bounds__(64) void softk_node(const float* __restrict__ x,
                                                 const float* __restrict__ ew,
                                                 const int*   __restrict__ nbr,
                                                 const float* __restrict__ bias,
                                                 float* __restrict__ out,
                                                 int n) {
  __shared__ NodeSmem sm[2];
  const int lane = threadIdx.x & 31;
  const int wave = threadIdx.x >> 5;
  const int node = blockIdx.x * 2 + wave;
  if (node >= n) return;
  NodeSmem& s = sm[wave];   // one wave per node: wave-synchronous LDS use

  // ---- load 49 candidates (48 edges + self-loop @ 1.0) ----
  for (int j = lane; j < DEG + 1; j += 32) {
    float wj; int nj;
    if (j < DEG) { wj = ew[(size_t)node * DEG + j]; nj = nbr[(size_t)node * DEG + j]; }
    else         { wj = 1.0f;                        nj = node; }
    s.wbuf[j]  = wj;
    s.nbbuf[j] = nj;
  }

  // ---- rank-based top-32 (descending, ties by lower index == lax.top_k) ----
  for (int j = lane; j < DEG + 1; j += 32) {
    float wj = s.wbuf[j];
    int rank = 0;
    for (int m = 0; m < DEG + 1; ++m) {
      float wm = s.wbuf[m];
      rank += (wm > wj) || (wm == wj && m < j);
    }
    if (rank < KSEL) { s.selw[rank] = wj; s.selidx[rank] = s.nbbuf[j]; }
  }

  // ---- gather xi = x[selidx] : 32 rows x 128 f32, coalesced float4 ----
  for (int a = 0; a < KSEL; ++a) {
    int row = s.selidx[a];
    float4 v = ((const float4*)(x + (size_t)row * DOUT))[lane];
    *(float4*)&s.xi[a * DOUT + lane * 4] = v;
  }

  // ---- Gram = xi @ xi^T : 4 WMMA tiles, K = 128 (32 steps of 4) ----
  // A-frag and B-frag per-lane patterns coincide for xi*xi^T: load two
  // fragments (rows 0-15, rows 16-31) and reuse each as both A and B.
  v8f g00 = {}, g01 = {}, g10 = {}, g11 = {};
  const int kg = (lane >> 4) << 1;
  const int r0 = (lane & 15) * DOUT + kg;
  const int r1 = (16 + (lane & 15)) * DOUT + kg;
#pragma unroll
  for (int kk = 0; kk < 32; ++kk) {
    v2f f0 = *(const v2f*)&s.xi[r0 + kk * 4];
    v2f f1 = *(const v2f*)&s.xi[r1 + kk * 4];
    g00 = wmma_f32(f0, f0, g00);
    g01 = wmma_f32(f0, f1, g01);
    g10 = wmma_f32(f1, f0, g10);
    g11 = wmma_f32(f1, f1, g11);
  }

  // ---- spill Gram tiles to LDS (C/D layout: VGPR r -> M=r / M=r+8) ----
  const int mlo = (lane >> 4) << 3;
  const int nlo = lane & 15;
#pragma unroll
  for (int r = 0; r < 8; ++r) {
    s.gram[(mlo + r) * KSEL + nlo]           = g00[r];
    s.gram[(mlo + r) * KSEL + 16 + nlo]      = g01[r];
    s.gram[(16 + mlo + r) * KSEL + nlo]      = g10[r];
    s.gram[(16 + mlo + r) * KSEL + 16 + nlo] = g11[r];
  }

  // ---- weighted medoid distances (sq comes from the Gram diagonal) ----
  float sa = s.gram[lane * KSEL + lane];
  float dist = 0.f;
  for (int b = 0; b < KSEL; ++b) {
    float g  = s.gram[lane * KSEL + b];
    float sb = s.gram[b * KSEL + b];
    float d2 = sa + sb - 2.f * g;
    float dm = (d2 > 0.f) ? sqrtf(d2) : 0.f;
    dist += s.selw[b] * dm;
  }
  s.distb[lane] = dist;

  // ---- rel = exp(-dist)*tw / sum(exp(-dist)*tw)  (softmax denom cancels) ----
  float mx = -3.402823466e38f;
  for (int b = 0; b < KSEL; ++b) mx = fmaxf(mx, -s.distb[b]);
  float num = expf(-dist - mx) * s.selw[lane];
  s.relb[lane] = num;
  float denom = 0.f;
  for (int b = 0; b < KSEL; ++b) denom += s.relb[b];
  float rel = num / denom;
  s.relb[lane] = rel;   // all reads of the numerators precede this (in-order DS)

  // ---- out[node] = rel^T @ xi + bias : lane owns a float4 column group ----
  float4 acc; acc.x = 0.f; acc.y = 0.f; acc.z = 0.f; acc.w = 0.f;
  for (int a = 0; a < KSEL; ++a) {
    float r = s.relb[a];
    float4 xa = *(const float4*)&s.xi[a * DOUT + lane * 4];
    acc.x = fmaf(r, xa.x, acc.x);
    acc.y = fmaf(r, xa.y, acc.y);
    acc.z = fmaf(r, xa.z, acc.z);
    acc.w = fmaf(r, xa.w, acc.w);
  }
  float4 b4 = ((const float4*)bias)[lane];
  acc.x += b4.x; acc.y += b4.y; acc.z += b4.z; acc.w += b4.w;
  ((float4*)(out + (size_t)node * DOUT))[lane] = acc;
}

// ---------------------------------------------------------------------------
// Launch
// ---------------------------------------------------------------------------
extern "C" void kernel_launch(void* const* d_in, const int* in_sizes, int n_in,
                              void* d_out, int out_size, void* d_ws, size_t ws_size,
                              hipStream_t stream) {
  const float* feat   = (const float*)d_in[0];   // [N,128] f32
  const float* edge_w = (const float*)d_in[1];   // [N,48]  f32
  const float* weight = (const float*)d_in[2];   // [128,128] f32
  const float* bias   = (const float*)d_in[3];   // [128] f32
  const int*   nbrs   = (const int*)d_in[4];     // [N,48] i32
  float*       outp   = (float*)d_out;           // [N,128] f32

  const int n = in_sizes[0] / DIN;               // 50000 (multiple of 16)
  float* x = (float*)d_ws;                       // [N,128] f32 scratch (25.6 MB)

  dim3 g1(n / 16, 2);
  xform_gemm<<<g1, 128, 0, stream>>>(feat, weight, x);

  softk_node<<<(n + 1) / 2, 64, 0, stream>>>(x, edge_w, nbrs, bias, outp, n);
}